// BoxIoU_45973329936551
// MI455X (gfx1250) — compile-verified
//
#include <hip/hip_runtime.h>

#define AS3 __attribute__((address_space(3)))

constexpr int kB1 = 2000;   // boxes in box1 per batch
constexpr int kB2 = 2000;   // boxes in box2 per batch
constexpr int kTI = 8;      // box1 rows per block
constexpr int kThreads = 256;
constexpr int kColsPerThread = 4;                 // one float4 store per row
constexpr int kJTile = kThreads * kColsPerThread; // 1024 columns per block

__global__ __launch_bounds__(kThreads) void BoxIoU_45973329936551_kernel(
    const float* __restrict__ box1, const float* __restrict__ box2,
    float* __restrict__ out) {
  __shared__ float  s_raw[kTI * 4];  // raw (xc, yc, w, h) rows, DMA target
  __shared__ float4 s_cor[kTI];      // derived (x1, y1, x2, y2)
  __shared__ float  s_a1[kTI];       // derived area

  const int tid = threadIdx.x;
  const int n  = blockIdx.z;
  const int i0 = blockIdx.y * kTI;
  const int j0 = blockIdx.x * kJTile + tid * kColsPerThread;

  // ---- Stage box1 tile (kTI rows x 16B) into LDS via the CDNA5 async
  // global->LDS DMA path (ASYNCcnt-tracked). GVS addressing:
  //   vdst holds the LDS address, saddr + vaddr the global one
  if (tid < kTI) {
    unsigned lds_off = (unsigned)(unsigned long long)(AS3 float*)s_raw + (unsigned)tid * 16u;
    unsigned v_off   = (unsigned)tid * 16u;
    unsigned long long gbase =
        (unsigned long long)(const void*)(box1 + ((size_t)n * kB1 + i0) * 4);
    asm volatile("global_load_async_to_lds_b128 %0, %1, %2"
                 :: "v"(lds_off), "v"(v_off), "s"(gbase)
                 : "memory");
  }
  asm volatile("s_wait_asynccnt 0" ::: "memory");
  __syncthreads();

  // ---- Convert raw rows to corners + area ONCE per block (lanes 0..kTI-1),
  // so the 256-thread hot loop below does zero redundant conversion work.
  if (tid < kTI) {
    float bx = s_raw[tid * 4 + 0], by = s_raw[tid * 4 + 1];
    float bw = s_raw[tid * 4 + 2], bh = s_raw[tid * 4 + 3];
    float hw = 0.5f * bw, hh = 0.5f * bh;
    s_cor[tid] = make_float4(bx - hw, by - hh, bx + hw, by + hh);
    s_a1[tid]  = bw * bh;
  }
  __syncthreads();

  if (j0 >= kB2) return;  // 2000 % 4 == 0 -> whole-float4 granularity, no ragged tail

  // ---- Per-thread box2 columns: 4 boxes, fully coalesced float4 loads.
  float x1b[4], y1b[4], x2b[4], y2b[4], a2[4];
  const float4* p2 = (const float4*)(box2 + ((size_t)n * kB2 + j0) * 4);
#pragma unroll
  for (int k = 0; k < 4; ++k) {
    float4 b = p2[k];
    float hw = 0.5f * b.z, hh = 0.5f * b.w;
    x1b[k] = b.x - hw; x2b[k] = b.x + hw;
    y1b[k] = b.y - hh; y2b[k] = b.y + hh;
    a2[k]  = b.z * b.w;
  }

  float* outp = out + ((size_t)n * kB1 + i0) * kB2 + j0;
#pragma unroll
  for (int ii = 0; ii < kTI; ++ii) {
    // Broadcast LDS reads (same address across lanes -> conflict-free),
    // issued on the DS pipe, overlapping VALU work.
    float4 c = s_cor[ii];
    float a1 = s_a1[ii];

    float r[4];
#pragma unroll
    for (int k = 0; k < 4; ++k) {
      float x0 = fmaxf(c.x, x1b[k]);
      float y0 = fmaxf(c.y, y1b[k]);
      float x1 = fminf(c.z, x2b[k]);
      float y1 = fminf(c.w, y2b[k]);
      float w = fmaxf(x1 - x0, 0.0f);
      float h = fmaxf(y1 - y0, 0.0f);
      float inter = w * h;
      float uni = (a1 + a2[k]) - inter;
      // Fast divide: v_rcp_f32 (TRANS pipe, co-executes with VALU) + 1 Newton step.
      float rc = __builtin_amdgcn_rcpf(uni);
      rc = __builtin_fmaf(__builtin_fmaf(-uni, rc, 1.0f), rc, rc);
      r[k] = inter * rc;
    }
    // Coalesced 16B store; row base is 16B-aligned (2000 floats * 4B = 8000B).
    *(float4*)(outp + (size_t)ii * kB2) = make_float4(r[0], r[1], r[2], r[3]);
  }
}

extern "C" void kernel_launch(void* const* d_in, const int* in_sizes, int n_in,
                              void* d_out, int out_size, void* d_ws, size_t ws_size,
                              hipStream_t stream) {
  const float* box1 = (const float*)d_in[0];
  const float* box2 = (const float*)d_in[1];
  float* out = (float*)d_out;

  int nBatch = in_sizes[0] / (kB1 * 4);  // 8 for the reference shapes
  dim3 grid((kB2 + kJTile - 1) / kJTile,  // 2 column stripes
            kB1 / kTI,                    // 250 row tiles
            nBatch);                      // 8 batches
  BoxIoU_45973329936551_kernel<<<grid, kThreads, 0, stream>>>(box1, box2, out);
}